// GraphTransformerEncoder_9509057593394
// MI455X (gfx1250) — compile-verified
//
#include <hip/hip_runtime.h>
#include <hip/hip_bf16.h>

#define GN 4096
#define GE 65536
#define GF 128
#define GH 256

typedef float v2f __attribute__((ext_vector_type(2)));
typedef float v8f __attribute__((ext_vector_type(8)));

// ---------------------------------------------------------------------------
// Generic fp32 WMMA GEMM:  out[M,256] = concat_k(segments) @ W[Ktot,256] + bias
// Each segment: base (+ optional per-row gather index), leading dim, K length.
// Wave tile: 32(M) x 64(N) -> each B fragment feeds 2 WMMAs (halved B traffic).
// K-step 4 via v_wmma_f32_16x16x4_f32.
// Block: 256 threads = 8 waves = 64(M) x 256(N). M must be a multiple of 64.
// ---------------------------------------------------------------------------
struct Seg {
    const float* base;
    const int*   gather;
    int          ld;
    int          klen;
};

__global__ __launch_bounds__(256) void wmma_gemm_f32(
    Seg s0, Seg s1, Seg s2,
    const float* __restrict__ W, const float* __restrict__ bias,
    float* __restrict__ out, int M, int do_relu)
{
    const int tid  = threadIdx.x;
    const int lane = tid & 31;
    const int wave = tid >> 5;
    const int wn   = wave & 3;        // N-tile group: 0..3 (64 cols each)
    const int wm   = wave >> 2;       // M super-tile: 0..1 (32 rows each)
    const int m0   = blockIdx.x * 64 + wm * 32;
    const int n0   = wn * 64;
    const int l16  = lane & 15;
    const int hi   = lane >> 4;       // 0: K=k,k+1   1: K=k+2,k+3
    const int row0 = m0 + l16;        // rows for A fragment 0
    const int row1 = m0 + 16 + l16;   // rows for A fragment 1

    v8f acc[2][4];
#pragma unroll
    for (int t = 0; t < 2; ++t)
#pragma unroll
        for (int j = 0; j < 4; ++j) acc[t][j] = (v8f)0.0f;

    Seg segs[3] = {s0, s1, s2};
    int kbase = 0;
#pragma unroll
    for (int s = 0; s < 3; ++s) {
        const Seg sg = segs[s];
        if (sg.klen == 0) continue;
        const int g0 = sg.gather ? sg.gather[row0] : row0;
        const int g1 = sg.gather ? sg.gather[row1] : row1;
        const float* aptr0 = sg.base + (size_t)g0 * sg.ld;
        const float* aptr1 = sg.base + (size_t)g1 * sg.ld;
        for (int k = 0; k < sg.klen; k += 4) {
            const int koff = k + (hi << 1);
            // A 16x4 fragments: rows m0+l16 / m0+16+l16, cols koff, koff+1
            v2f a0 = *(const v2f*)(aptr0 + koff);
            v2f a1 = *(const v2f*)(aptr1 + koff);
            const float* wp = W + (size_t)(kbase + koff) * GH;
            // B 4x16 fragments: rows koff, koff+1, col = n0 + j*16 + l16
            v2f b[4];
#pragma unroll
            for (int j = 0; j < 4; ++j) {
                const int colj = n0 + j * 16 + l16;
                b[j].x = wp[colj];
                b[j].y = wp[GH + colj];
            }
#pragma unroll
            for (int j = 0; j < 4; ++j) {
                acc[0][j] = __builtin_amdgcn_wmma_f32_16x16x4_f32(
                    false, a0, false, b[j], (short)0, acc[0][j], false, false);
                acc[1][j] = __builtin_amdgcn_wmma_f32_16x16x4_f32(
                    false, a1, false, b[j], (short)0, acc[1][j], false, false);
            }
        }
        kbase += sg.klen;
    }

    // C/D layout: VGPR r -> M = mtile + r + 8*hi, N = n0tile + l16
#pragma unroll
    for (int t = 0; t < 2; ++t) {
        const int rbase = m0 + t * 16 + (hi << 3);
#pragma unroll
        for (int j = 0; j < 4; ++j) {
            const int colj = n0 + j * 16 + l16;
            const float bv = bias[colj];
#pragma unroll
            for (int r = 0; r < 8; ++r) {
                float v = acc[t][j][r] + bv;
                if (do_relu) v = fmaxf(v, 0.0f);
                out[(size_t)(rbase + r) * GH + colj] = v;
            }
        }
    }
}

// ---------------------------------------------------------------------------
// Helper kernels
// ---------------------------------------------------------------------------
__global__ __launch_bounds__(256) void zero_u32(unsigned* __restrict__ p, size_t n)
{
    size_t i = (size_t)blockIdx.x * blockDim.x + threadIdx.x;
    size_t stride = (size_t)gridDim.x * blockDim.x;
    for (; i < n; i += stride) p[i] = 0u;
}

// nsum[row[e]] += node[col[e]]; deg[row[e]] += 1   (16 edges per block)
__global__ __launch_bounds__(256) void scatter_mean1(
    const int* __restrict__ row, const int* __restrict__ col,
    const float* __restrict__ node, float* __restrict__ nsum,
    float* __restrict__ deg)
{
    const int t  = threadIdx.x;
    const int e0 = blockIdx.x * 16;
#pragma unroll
    for (int i = 0; i < 16; ++i) {
        const int e = e0 + i;
        const int r = row[e];
        const int c = col[e];
        const float v = node[(size_t)c * GH + t];
        atomicAdd(&nsum[(size_t)r * GH + t], v);
        if (t == 0) atomicAdd(&deg[r], 1.0f);
    }
}

__global__ __launch_bounds__(256) void finalize_mean1(
    float* __restrict__ mean1, const float* __restrict__ deg)
{
    const int i = blockIdx.x;
    const float d = deg[i];
    const size_t o = (size_t)i * GH + threadIdx.x;
    mean1[o] = (d > 0.0f) ? mean1[o] / d : 0.0f;
}

// adj[row][col] bit set (1-hop adjacency bitsets, 128 words per node)
__global__ __launch_bounds__(256) void build_adj(
    const int* __restrict__ row, const int* __restrict__ col,
    unsigned* __restrict__ adj)
{
    const int e = blockIdx.x * 256 + threadIdx.x;
    if (e < GE) {
        const int r = row[e];
        const int c = col[e];
        atomicOr(&adj[(size_t)r * 128 + (c >> 5)], 1u << (c & 31));
    }
}

// bits2[row[e]] |= adj[col[e]]  (one edge per block, 128 words)
__global__ __launch_bounds__(128) void or_twohop(
    const int* __restrict__ row, const int* __restrict__ col,
    const unsigned* __restrict__ adj, unsigned* __restrict__ bits2)
{
    const int e = blockIdx.x;
    const int r = row[e];
    const int c = col[e];
    const unsigned a = adj[(size_t)c * 128 + threadIdx.x];
    if (a) atomicOr(&bits2[(size_t)r * 128 + threadIdx.x], a);
}

// mean2[i] = mean over set bits j (j != i) of node[j]
__global__ __launch_bounds__(256) void mean2_kernel(
    const unsigned* __restrict__ bits2, const float* __restrict__ node,
    float* __restrict__ mean2)
{
    __shared__ unsigned sb[128];
    __shared__ int pcs[128];
    const int i = blockIdx.x;
    const int t = threadIdx.x;

    if (t < 128) {
        unsigned w = bits2[(size_t)i * 128 + t];
        if (t == (i >> 5)) w &= ~(1u << (i & 31));   // remove self
        sb[t]  = w;
        pcs[t] = __popc(w);
    }
    __syncthreads();
    for (int s = 64; s > 0; s >>= 1) {
        if (t < s) pcs[t] += pcs[t + s];
        __syncthreads();
    }
    const int cnt = pcs[0];

    float acc = 0.0f;
    for (int wd = 0; wd < 128; ++wd) {
        unsigned bits = sb[wd];
        while (bits) {
            const int b = __ffs(bits) - 1;
            bits &= bits - 1;
            acc += node[(size_t)(wd * 32 + b) * GH + t];
        }
    }
    mean2[(size_t)i * GH + t] = (cnt > 0) ? acc / (float)cnt : 0.0f;
}

// ---------------------------------------------------------------------------
// Launch
// ---------------------------------------------------------------------------
extern "C" void kernel_launch(void* const* d_in, const int* in_sizes, int n_in,
                              void* d_out, int out_size, void* d_ws, size_t ws_size,
                              hipStream_t stream)
{
    const float* x   = (const float*)d_in[0];
    const int*   ei  = (const int*)d_in[1];
    const float* Wn  = (const float*)d_in[2];
    const float* bn  = (const float*)d_in[3];
    const float* We  = (const float*)d_in[4];
    const float* be  = (const float*)d_in[5];
    const float* Ws1 = (const float*)d_in[6];
    const float* bs1 = (const float*)d_in[7];
    const float* Ws2 = (const float*)d_in[8];
    const float* bs2 = (const float*)d_in[9];
    const float* Wh1 = (const float*)d_in[10];
    const float* bh1 = (const float*)d_in[11];
    const float* Wh2 = (const float*)d_in[12];
    const float* bh2 = (const float*)d_in[13];

    const int* row = ei;        // edge_index[0]
    const int* col = ei + GE;   // edge_index[1]

    float* node_out = (float*)d_out;
    float* edge_out = node_out + (size_t)GN * GH;
    float* sub_out  = edge_out + (size_t)GE * GH;
    float* nbh_out  = sub_out  + (size_t)GN * GH;

    // workspace layout (mean1 | deg | adj | bits2 zeroed together; mean2/h1 overwritten)
    float*    mean1 = (float*)d_ws;                       // GN*GH
    float*    deg   = mean1 + (size_t)GN * GH;            // GN
    unsigned* adj   = (unsigned*)(deg + GN);              // GN*128
    unsigned* bits2 = adj + (size_t)GN * 128;             // GN*128
    float*    mean2 = (float*)(bits2 + (size_t)GN * 128); // GN*GH
    float*    h1    = mean2 + (size_t)GN * GH;            // GN*GH (reused by both MLPs)

    const size_t zero_words = (size_t)GN * GH + GN + (size_t)GN * 128 * 2;
    zero_u32<<<4096, 256, 0, stream>>>((unsigned*)d_ws, zero_words);

    const Seg nil = {nullptr, nullptr, 0, 0};

    // node = x @ Wn + bn
    wmma_gemm_f32<<<GN / 64, 256, 0, stream>>>(
        Seg{x, nullptr, GF, GF}, nil, nil, Wn, bn, node_out, GN, 0);

    // edge = [x[row] | x[col]] @ We + be
    wmma_gemm_f32<<<GE / 64, 256, 0, stream>>>(
        Seg{x, row, GF, GF}, Seg{x, col, GF, GF}, nil, We, be, edge_out, GE, 0);

    // mean1 = segment_mean(node[col], row)
    scatter_mean1<<<GE / 16, 256, 0, stream>>>(row, col, node_out, mean1, deg);
    finalize_mean1<<<GN, 256, 0, stream>>>(mean1, deg);

    // two-hop reachability bitsets, then mean2
    build_adj<<<GE / 256, 256, 0, stream>>>(row, col, adj);
    or_twohop<<<GE, 128, 0, stream>>>(row, col, adj, bits2);
    mean2_kernel<<<GN, 256, 0, stream>>>(bits2, node_out, mean2);

    // sub = relu([node|mean1] @ Ws1 + bs1) @ Ws2 + bs2
    wmma_gemm_f32<<<GN / 64, 256, 0, stream>>>(
        Seg{node_out, nullptr, GH, GH}, Seg{mean1, nullptr, GH, GH}, nil,
        Ws1, bs1, h1, GN, 1);
    wmma_gemm_f32<<<GN / 64, 256, 0, stream>>>(
        Seg{h1, nullptr, GH, GH}, nil, nil, Ws2, bs2, sub_out, GN, 0);

    // nbh = relu([node|mean1|mean2] @ Wh1 + bh1) @ Wh2 + bh2
    wmma_gemm_f32<<<GN / 64, 256, 0, stream>>>(
        Seg{node_out, nullptr, GH, GH}, Seg{mean1, nullptr, GH, GH},
        Seg{mean2, nullptr, GH, GH}, Wh1, bh1, h1, GN, 1);
    wmma_gemm_f32<<<GN / 64, 256, 0, stream>>>(
        Seg{h1, nullptr, GH, GH}, nil, nil, Wh2, bh2, nbh_out, GN, 0);
}